// EGGTRM_7954279432813
// MI455X (gfx1250) — compile-verified
//
#include <hip/hip_runtime.h>

typedef __attribute__((ext_vector_type(2))) float v2f;
typedef __attribute__((ext_vector_type(8))) float v8f;

namespace {
constexpr int BATCH = 262144;
constexpr int HID   = 64;
constexpr int OUTD  = 5;
constexpr int LCYC  = 6;
constexpr int HCYC  = 3;

constexpr int ROWS_PER_WAVE = 16;
constexpr int WAVES_PER_WG  = 8;
constexpr int ROWS_PER_WG   = ROWS_PER_WAVE * WAVES_PER_WG;  // 128
constexpr int NBLK          = BATCH / ROWS_PER_WG;           // 2048

// Weights live in LDS in B-fragment-major layout:
//   frag = kk*(N/16) + nt ; float index = (frag*32 + lane)*2 + j
// so a B fragment is one contiguous ds_load_b64 per lane, conflict-free.
constexpr int OFF_WIN = 0;                    // 64x64   -> 4096
constexpr int OFF_WUP = OFF_WIN + 64 * 64;    // 64x256  -> 16384
constexpr int OFF_WDN = OFF_WUP + 64 * 256;   // 256x64  -> 16384
constexpr int OFF_WCB = OFF_WDN + 256 * 64;   // 128x64  -> 8192
constexpr int OFF_WTR = OFF_WCB + 128 * 64;   // 64x64   -> 4096
constexpr int OFF_WHD = OFF_WTR + 64 * 64;    // 64x16   -> 1024
constexpr int OFF_SCR = OFF_WHD + 64 * 16;

constexpr int SCR_LD = 68;                    // padded: A-loads/C-stores conflict-free
constexpr int SCR_PER_WAVE = 2 * 16 * SCR_LD; // scrZ + scrU
constexpr int OFF_CN = OFF_SCR + WAVES_PER_WG * SCR_PER_WAVE;  // folded consts
constexpr int OFF_CT = OFF_CN + 64;
constexpr int LDS_FLOATS = OFF_CT + 64;
constexpr int LDS_BYTES  = LDS_FLOATS * 4;    // ~264 KB <= 320 KB/WGP
}

// D = A(16x4 f32) * B(4x16 f32) + C(16x16 f32)
__device__ __forceinline__ v8f wmma4(v2f a, v2f b, v8f c) {
  return __builtin_amdgcn_wmma_f32_16x16x4_f32(false, a, false, b, (short)0, c,
                                               false, false);
}

// Intra-wave LDS ordering. Hardware processes a wave's DS ops in order
// (ISA ch11 §6),
// so a store->load roundtrip to per-wave scratch needs no
// explicit s_wait_dscnt: the compiler-inserted wait on the load's own result
// transitively covers the preceding stores. We only need to stop the
// compiler from reordering LDS ops across this point.
__device__ __forceinline__ void lds_order() {
  asm volatile("" ::: "memory");
}

// contiguous B fragment (pre-swizzled): one b64 per lane
__device__ __forceinline__ v2f ldBf(const float* Wf, int frag, int lane) {
  return *(const v2f*)(Wf + frag * 64 + lane * 2);
}

// A fragment from row-major [16][SCR_LD] tile: element (m,k) -> lane ((k-k0)>>1)*16+m
__device__ __forceinline__ v2f ldA(const float* S, int k0, int g, int ln) {
  return *(const v2f*)(S + ln * SCR_LD + k0 + 2 * g);
}

// C/D fragment store: vgpr v, lane (g*16+n) -> element (8g+v, n0+n)
__device__ __forceinline__ void stC(float* S, int n0, int g, int ln, v8f c) {
  float* p = S + (8 * g) * SCR_LD + n0 + ln;
#pragma unroll
  for (int v = 0; v < 8; ++v) p[v * SCR_LD] = c[v];
}

// stage W (K x N row-major, N multiple of 16) into fragment-major LDS layout
__device__ __forceinline__ void stageB(float* dst, const float* src, int K,
                                       int N, int tid) {
  const int NT = N >> 4;
  for (int i = tid; i < K * N; i += 256) {
    int j    = i & 1;
    int lane = (i >> 1) & 31;
    int rest = i >> 6;            // kk*NT + nt
    int nt   = rest % NT;
    int kk   = rest / NT;
    int k = kk * 4 + 2 * (lane >> 4) + j;
    int n = nt * 16 + (lane & 15);
    dst[i] = src[k * N + n];
  }
}

__global__ __launch_bounds__(256, 1)
void hrm_fused(const float* __restrict__ x,
               const float* __restrict__ W_in,   const float* __restrict__ b_in,
               const float* __restrict__ y_init,
               const float* __restrict__ W_up,   const float* __restrict__ b_up,
               const float* __restrict__ W_down, const float* __restrict__ b_down,
               const float* __restrict__ W_comb, const float* __restrict__ b_comb,
               const float* __restrict__ W_trans,const float* __restrict__ b_trans,
               const float* __restrict__ W_head, const float* __restrict__ b_head,
               float* __restrict__ out) {
  extern __shared__ float lds[];
  const int tid  = threadIdx.x;
  const int wave = tid >> 5;
  const int lane = tid & 31;
  const int g    = lane >> 4;
  const int ln   = lane & 15;

  // ---- stage all weights into LDS, pre-swizzled to fragment layout ----
  stageB(lds + OFF_WIN, W_in,   64,  64, tid);
  stageB(lds + OFF_WUP, W_up,   64, 256, tid);
  stageB(lds + OFF_WDN, W_down, 256, 64, tid);
  stageB(lds + OFF_WCB, W_comb, 128, 64, tid);
  stageB(lds + OFF_WTR, W_trans, 64, 64, tid);
  for (int i = tid; i < 64 * 16; i += 256) {  // W_head zero-padded to 64x16
    int j = i & 1, ll = (i >> 1) & 31, kk = i >> 6;
    int k = kk * 4 + 2 * (ll >> 4) + j;
    int n = ll & 15;
    lds[OFF_WHD + i] = (n < OUTD) ? W_head[k * OUTD + n] : 0.f;
  }

  // ---- folded affine constants, computed once per workgroup ----
  // cdn = b_down + b_up @ W_down ; ctr = b_trans + b_comb @ W_trans
  if (tid < 64) {
    float s = b_down[tid];
#pragma unroll 4
    for (int k = 0; k < 256; ++k) s += b_up[k] * W_down[k * 64 + tid];
    lds[OFF_CN + tid] = s;
  } else if (tid < 128) {
    const int c2 = tid - 64;
    float s = b_trans[c2];
#pragma unroll 4
    for (int k = 0; k < 64; ++k) s += b_comb[k] * W_trans[k * 64 + c2];
    lds[OFF_CT + c2] = s;
  }

  __syncthreads();   // the only cross-wave dependency in the kernel

  const float* WIN = lds + OFF_WIN;
  const float* WUP = lds + OFF_WUP;
  const float* WDN = lds + OFF_WDN;
  const float* WCB = lds + OFF_WCB;
  const float* WTR = lds + OFF_WTR;
  const float* WHD = lds + OFF_WHD;
  float* scrZ = lds + OFF_SCR + wave * SCR_PER_WAVE;
  float* scrU = scrZ + 16 * SCR_LD;

  const int row0 = blockIdx.x * ROWS_PER_WG + wave * ROWS_PER_WAVE;
  const v8f vzero{};

  // per-lane constants (column index = nt*16+ln)
  float bin[4], yin[4], cdn[4], ctr[4];
#pragma unroll
  for (int nt = 0; nt < 4; ++nt) {
    const int col = nt * 16 + ln;
    bin[nt] = b_in[col];
    yin[nt] = y_init[col];
    cdn[nt] = lds[OFF_CN + col];
    ctr[nt] = lds[OFF_CT + col];
  }
  const float bhd = (ln < OUTD) ? b_head[ln] : 0.f;

  // ---- z = x @ W_in + b_in  (A frags straight from global: row m = ln) ----
  v8f z[4];
  {
    const float* xr = x + (long)(row0 + ln) * HID;
    {
      v2f a = *(const v2f*)(xr + 2 * g);
#pragma unroll
      for (int nt = 0; nt < 4; ++nt)
        z[nt] = wmma4(a, ldBf(WIN, nt, lane), vzero);   // C = inline 0
    }
#pragma unroll
    for (int kk = 1; kk < 16; ++kk) {
      v2f a = *(const v2f*)(xr + kk * 4 + 2 * g);
#pragma unroll
      for (int nt = 0; nt < 4; ++nt)
        z[nt] = wmma4(a, ldBf(WIN, kk * 4 + nt, lane), z[nt]);
    }
#pragma unroll
    for (int nt = 0; nt < 4; ++nt) {
      v8f t = z[nt];
#pragma unroll
      for (int v = 0; v < 8; ++v) t[v] += bin[nt];
      z[nt] = t;
    }
  }

  // ---- y = broadcast(y_init) ----
  v8f y[4];
#pragma unroll
  for (int nt = 0; nt < 4; ++nt) {
    v8f t;
#pragma unroll
    for (int v = 0; v < 8; ++v) t[v] = yin[nt];
    y[nt] = t;
  }

#pragma unroll 1
  for (int h = 0; h < HCYC; ++h) {
#pragma unroll 1
    for (int l = 0; l < LCYC; ++l) {
      lds_order();
#pragma unroll
      for (int nt = 0; nt < 4; ++nt) stC(scrZ, nt * 16, g, ln, z[nt]);
      lds_order();
      v2f za[16];                        // keep all z A-frags in registers
#pragma unroll
      for (int kk = 0; kk < 16; ++kk) za[kk] = ldA(scrZ, kk * 4, g, ln);

      v8f zn[4];                         // zn = z + (b_up@W_down + b_down)
#pragma unroll
      for (int nt = 0; nt < 4; ++nt) {
        v8f t = z[nt];
#pragma unroll
        for (int v = 0; v < 8; ++v) t[v] += cdn[nt];
        zn[nt] = t;
      }
#pragma unroll 1
      for (int c = 0; c < 4; ++c) {      // 4 column-chunks of the 16x256 up-proj
        v8f u[4];
#pragma unroll
        for (int nt = 0; nt < 4; ++nt)   // first K-step: C = inline 0
          u[nt] = wmma4(za[0], ldBf(WUP, c * 4 + nt, lane), vzero);
#pragma unroll
        for (int kk = 1; kk < 16; ++kk) {
#pragma unroll
          for (int nt = 0; nt < 4; ++nt)
            u[nt] = wmma4(za[kk], ldBf(WUP, kk * 16 + c * 4 + nt, lane), u[nt]);
        }
        lds_order();
#pragma unroll
        for (int nt = 0; nt < 4; ++nt) stC(scrU, nt * 16, g, ln, u[nt]);
        lds_order();
#pragma unroll
        for (int kk = 0; kk < 16; ++kk) {
          v2f a = ldA(scrU, kk * 4, g, ln);
#pragma unroll
          for (int nt = 0; nt < 4; ++nt)
            zn[nt] = wmma4(a, ldBf(WDN, (c * 16 + kk) * 4 + nt, lane), zn[nt]);
        }
      }
#pragma unroll
      for (int nt = 0; nt < 4; ++nt) z[nt] = zn[nt];
    }

    // ---- y += (concat(z,y) @ W_comb + b_comb) @ W_trans + b_trans ----
    lds_order();
#pragma unroll
    for (int nt = 0; nt < 4; ++nt) {
      stC(scrZ, nt * 16, g, ln, z[nt]);
      stC(scrU, nt * 16, g, ln, y[nt]);
    }
    lds_order();
    v8f t4[4];
    {
      v2f a = ldA(scrZ, 0, g, ln);
#pragma unroll
      for (int nt = 0; nt < 4; ++nt)
        t4[nt] = wmma4(a, ldBf(WCB, nt, lane), vzero);  // C = inline 0
    }
#pragma unroll
    for (int kk = 1; kk < 16; ++kk) {    // K = 4..63 from z
      v2f a = ldA(scrZ, kk * 4, g, ln);
#pragma unroll
      for (int nt = 0; nt < 4; ++nt)
        t4[nt] = wmma4(a, ldBf(WCB, kk * 4 + nt, lane), t4[nt]);
    }
#pragma unroll
    for (int kk = 0; kk < 16; ++kk) {    // K = 64..127 from y
      v2f a = ldA(scrU, kk * 4, g, ln);
#pragma unroll
      for (int nt = 0; nt < 4; ++nt)
        t4[nt] = wmma4(a, ldBf(WCB, (16 + kk) * 4 + nt, lane), t4[nt]);
    }
    lds_order();
#pragma unroll
    for (int nt = 0; nt < 4; ++nt) stC(scrZ, nt * 16, g, ln, t4[nt]);
    lds_order();
#pragma unroll
    for (int nt = 0; nt < 4; ++nt) {     // y += (b_comb@W_trans + b_trans)
      v8f t = y[nt];
#pragma unroll
      for (int v = 0; v < 8; ++v) t[v] += ctr[nt];
      y[nt] = t;
    }
#pragma unroll
    for (int kk = 0; kk < 16; ++kk) {
      v2f a = ldA(scrZ, kk * 4, g, ln);
#pragma unroll
      for (int nt = 0; nt < 4; ++nt)
        y[nt] = wmma4(a, ldBf(WTR, kk * 4 + nt, lane), y[nt]);
    }
  }

  // ---- out = y @ W_head + b_head (padded 64x16 tile, store 5 cols) ----
  lds_order();
#pragma unroll
  for (int nt = 0; nt < 4; ++nt) stC(scrZ, nt * 16, g, ln, y[nt]);
  lds_order();
  v8f o = wmma4(ldA(scrZ, 0, g, ln), ldBf(WHD, 0, lane), vzero);
#pragma unroll
  for (int kk = 1; kk < 16; ++kk) {
    v2f a = ldA(scrZ, kk * 4, g, ln);
    o = wmma4(a, ldBf(WHD, kk, lane), o);
  }
  if (ln < OUTD) {
#pragma unroll
    for (int v = 0; v < 8; ++v)
      out[(long)(row0 + 8 * g + v) * OUTD + ln] = o[v] + bhd;
  }
}

extern "C" void kernel_launch(void* const* d_in, const int* in_sizes, int n_in,
                              void* d_out, int out_size, void* d_ws, size_t ws_size,
                              hipStream_t stream) {
  const float* x       = (const float*)d_in[0];
  const float* W_in    = (const float*)d_in[1];
  const float* b_in    = (const float*)d_in[2];
  const float* y_init  = (const float*)d_in[3];
  const float* W_up    = (const float*)d_in[4];
  const float* b_up    = (const float*)d_in[5];
  const float* W_down  = (const float*)d_in[6];
  const float* b_down  = (const float*)d_in[7];
  const float* W_comb  = (const float*)d_in[8];
  const float* b_comb  = (const float*)d_in[9];
  const float* W_trans = (const float*)d_in[10];
  const float* b_trans = (const float*)d_in[11];
  const float* W_head  = (const float*)d_in[12];
  const float* b_head  = (const float*)d_in[13];
  float* out = (float*)d_out;

  (void)hipFuncSetAttribute((const void*)hrm_fused,
                            hipFuncAttributeMaxDynamicSharedMemorySize, LDS_BYTES);
  hipLaunchKernelGGL(hrm_fused, dim3(NBLK), dim3(256), LDS_BYTES, stream,
                     x, W_in, b_in, y_init, W_up, b_up, W_down, b_down,
                     W_comb, b_comb, W_trans, b_trans, W_head, b_head, out);
}